// AutoregressiveDecoder_42167988912183
// MI455X (gfx1250) — compile-verified
//
#include <hip/hip_runtime.h>
#include <math.h>

typedef float v8f __attribute__((ext_vector_type(8)));
typedef float v2f __attribute__((ext_vector_type(2)));

__constant__ int c_perms[6][3] = {{0,1,2},{0,2,1},{1,0,2},{1,2,0},{2,0,1},{2,1,0}};

__device__ __forceinline__ float gelu_exact(float x) {
    return 0.5f * x * (1.0f + erff(x * 0.70710678118654752440f));
}

// W is stored K-pair interleaved: wpack[k2*WP_STRIDE + 2n + {0,1}] = W[2k2][n], W[2k2+1][n]
// WP_STRIDE = 288 floats -> 288 % 64 == 32 banks: lanes 0-15 vs 16-31 hit disjoint bank halves.
#define WP_STRIDE 288
#define ASTRIDE   132   // activation LDS row stride; 132 % 64 == 4 -> conflict-free A b64 frags

// pack one 128x128 row-major chunk of W into the K-pair-interleaved LDS layout
__device__ __forceinline__ void pack_w_chunk(float* wpack, const float* Wsrc, int tid) {
#pragma unroll
    for (int it = 0; it < 8; ++it) {
        int u  = it * 256 + tid;     // 2048 units = 64 k2-rows x 32 n4-units
        int k2 = u >> 5;
        int n  = (u & 31) * 4;
        float4 a0 = *(const float4*)(Wsrc + (size_t)(2 * k2)     * 128 + n);
        float4 a1 = *(const float4*)(Wsrc + (size_t)(2 * k2 + 1) * 128 + n);
        float4 p0 = make_float4(a0.x, a1.x, a0.y, a1.y);
        float4 p1 = make_float4(a0.z, a1.z, a0.w, a1.w);
        float* dst = wpack + (size_t)k2 * WP_STRIDE + 2 * n;
        *(float4*)(dst)     = p0;
        *(float4*)(dst + 4) = p1;
    }
}

__global__ __launch_bounds__(256) void ar_decoder_kernel(
    const float* __restrict__ seq,   const float* __restrict__ freq,
    const float* __restrict__ pres,  const float* __restrict__ rmask,
    const int*   __restrict__ perm_idx,
    const float* __restrict__ W1,    const float* __restrict__ b1,
    const float* __restrict__ W2,    const float* __restrict__ b2,
    const float* __restrict__ W3,    const float* __restrict__ b3,
    float* __restrict__ out, float* __restrict__ ws, int Btot)
{
    extern __shared__ float smem[];
    float* wpack    = smem;                       // 64 x WP_STRIDE (W1 k-chunk, later W2)
    float* abuf     = wpack + 64 * WP_STRIDE;     // 8 waves x 16 x ASTRIDE (seq chunk / h1 / h2)
    float* w1tail   = abuf + 8 * 16 * ASTRIDE;    // 12 x 128 (W1 rows 1024..1035)
    float* statebuf = w1tail + 12 * 128;          // 8 waves x 16 rows x 12 floats

    const int tid   = threadIdx.x;
    const int wv    = tid >> 5;
    const int lane  = tid & 31;
    const int lrow  = lane & 15;
    const int lhalf = lane >> 4;
    const int growbase = blockIdx.x * 128 + wv * 16;

    float* abw = abuf + wv * 16 * ASTRIDE;
    float* sbw = statebuf + wv * 16 * 12;

    // base accumulator = b1 broadcast (C layout: VGPR r, lane l -> row r+8*lhalf, col nt*16+lrow)
    v8f acc[8];
#pragma unroll
    for (int nt = 0; nt < 8; ++nt) {
        float bv = b1[nt * 16 + lrow];
#pragma unroll
        for (int r = 0; r < 8; ++r) acc[nt][r] = bv;
    }

    // ---------------- Phase 1: base = seq_embed @ W1[0:1024] (+b1), K chunked by 128 ----------
    for (int c = 0; c < 8; ++c) {
        // cooperative W1 chunk -> wpack (K-pair interleaved)
        pack_w_chunk(wpack, W1 + (size_t)c * 128 * 128, tid);
        // per-wave seq chunk -> abw (coalesced float4 per row)
#pragma unroll
        for (int i = 0; i < 16; ++i) {
            *(float4*)(abw + i * ASTRIDE + lane * 4) =
                *(const float4*)(seq + (size_t)(growbase + i) * 1024 + c * 128 + lane * 4);
        }
        // prefetch next chunk of W1 + seq so HBM overlaps this chunk's WMMAs
        if (c < 7) {
            const float* nw = W1 + (size_t)(c + 1) * 128 * 128;
            __builtin_prefetch(nw + tid * 64, 0, 1);
            __builtin_prefetch(nw + tid * 64 + 32, 0, 1);
            const float* nsq = seq + (size_t)(growbase + lrow) * 1024 + (c + 1) * 128 + lhalf * 64;
            __builtin_prefetch(nsq, 0, 1);
            __builtin_prefetch(nsq + 32, 0, 1);
        }
        __syncthreads();

        for (int k = 0; k < 32; ++k) {
            int k4 = k * 4;
            // A 16x4 f32 frag: lane l -> row l%16; v0 = K=k4+2*lhalf, v1 = K=k4+1+2*lhalf
            v2f a = *(const v2f*)(abw + lrow * ASTRIDE + k4 + 2 * lhalf);
#pragma unroll
            for (int nt = 0; nt < 8; ++nt) {
                // contiguous b64: (W[k4+2lh][col], W[k4+2lh+1][col])
                v2f b = *(const v2f*)(wpack + (size_t)(2 * k + lhalf) * WP_STRIDE
                                      + 2 * (nt * 16 + lrow));
                acc[nt] = __builtin_amdgcn_wmma_f32_16x16x4_f32(
                    false, a, false, b, (short)0, acc[nt], false, false);
            }
        }
        __syncthreads();
    }

    // ---------------- Phase 2 prologue: W2 -> wpack, W1 tail -> LDS, state init ----------
    pack_w_chunk(wpack, W2, tid);
    for (int idx = tid * 4; idx < 12 * 128; idx += 1024)
        *(float4*)(w1tail + idx) = *(const float4*)(W1 + (size_t)1024 * 128 + idx);

    float fr[3], pr[3], rm[3];
    if (lane < 16) {
        int grow = growbase + lane;
#pragma unroll
        for (int j = 0; j < 9; ++j) sbw[lane * 12 + j] = 0.0f;
        int pi = perm_idx[grow];
#pragma unroll
        for (int s = 0; s < 3; ++s) sbw[lane * 12 + 9 + s] = (float)c_perms[pi][s];
#pragma unroll
        for (int j = 0; j < 3; ++j) {
            fr[j] = freq[grow * 3 + j];
            pr[j] = pres[grow * 3 + j];
            rm[j] = rmask[grow * 3 + j];
        }
    }
    __syncthreads();

    float lloss = 0.0f, lm = 0.0f;

    // ---------------- 3 recurrent steps (wave-local) ----------------
    for (int s = 0; s < 3; ++s) {
        // h1 = gelu(base + state@W1tail + W1[1033+ridx]) computed in C layout, stored A-major
#pragma unroll
        for (int r = 0; r < 8; ++r) {
            int row = r + 8 * lhalf;
            float st[9];
#pragma unroll
            for (int j = 0; j < 9; ++j) st[j] = sbw[row * 12 + j];
            int ridx = (int)sbw[row * 12 + 9 + s];
            const float* woh = w1tail + (9 + ridx) * 128;
#pragma unroll
            for (int nt = 0; nt < 8; ++nt) {
                int col = nt * 16 + lrow;
                float d = woh[col];
#pragma unroll
                for (int j = 0; j < 9; ++j) d += st[j] * w1tail[j * 128 + col];
                abw[row * ASTRIDE + col] = gelu_exact(acc[nt][r] + d);
            }
        }

        // h2 = gelu(h1 @ W2 + b2) via WMMA (K=128)
        v8f acc2[8];
#pragma unroll
        for (int nt = 0; nt < 8; ++nt) {
            float bv = b2[nt * 16 + lrow];
#pragma unroll
            for (int r = 0; r < 8; ++r) acc2[nt][r] = bv;
        }
        for (int k = 0; k < 32; ++k) {
            int k4 = k * 4;
            v2f a = *(const v2f*)(abw + lrow * ASTRIDE + k4 + 2 * lhalf);
#pragma unroll
            for (int nt = 0; nt < 8; ++nt) {
                v2f b = *(const v2f*)(wpack + (size_t)(2 * k + lhalf) * WP_STRIDE
                                      + 2 * (nt * 16 + lrow));
                acc2[nt] = __builtin_amdgcn_wmma_f32_16x16x4_f32(
                    false, a, false, b, (short)0, acc2[nt], false, false);
            }
        }
#pragma unroll
        for (int nt = 0; nt < 8; ++nt)
#pragma unroll
            for (int r = 0; r < 8; ++r) {
                int row = r + 8 * lhalf;
                abw[row * ASTRIDE + nt * 16 + lrow] = gelu_exact(acc2[nt][r]);
            }

        // pred = h2 @ W3 + b3 : 2 lanes per row, each sums 64 cols
        float pf = 0.0f, z = 0.0f;
        const float* h2row = abw + lrow * ASTRIDE + lhalf * 64;
#pragma unroll
        for (int n = 0; n < 64; ++n) {
            float hv = h2row[n];
            int gn = lhalf * 64 + n;
            pf += hv * W3[gn * 2 + 0];
            z  += hv * W3[gn * 2 + 1];
        }
        pf += __shfl_xor(pf, 16);
        z  += __shfl_xor(z, 16);
        pf += b3[0];
        z  += b3[1];

        if (lane < 16) {
            int grow = growbase + lane;
            int ridx = (int)sbw[lane * 12 + 9 + s];
            float gf = fr[ridx], gp = pr[ridx], m = rm[ridx];
            float diff = pf - gf;
            float fl = diff * diff * m;
            float pl = (fmaxf(z, 0.0f) - z * gp + log1pf(expf(-fabsf(z)))) * m;
            lloss += fl + pl;
            lm += m;
            float af = (m > 0.5f) ? pf : gf;
            float ap = (m > 0.5f) ? (1.0f / (1.0f + expf(-z))) : gp;
            sbw[lane * 12 + ridx * 3 + 0] = af;
            sbw[lane * 12 + ridx * 3 + 1] = ap;
            sbw[lane * 12 + ridx * 3 + 2] = 1.0f;
            out[1 + (size_t)grow * 3 + ridx] = af;
            out[1 + (size_t)Btot * 3 + (size_t)grow * 3 + ridx] = ap;
        }
    }

    // reduce loss over the 16 owner lanes (lanes 16..31 hold zeros)
#pragma unroll
    for (int off = 8; off >= 1; off >>= 1) {
        lloss += __shfl_xor(lloss, off);
        lm    += __shfl_xor(lm, off);
    }
    if (lane == 0) {
        atomicAdd(ws + 0, lloss);
        atomicAdd(ws + 1, lm);
    }
}

__global__ void ar_init_kernel(float* ws) {
    ws[0] = 0.0f;
    ws[1] = 0.0f;
}

__global__ void ar_fin_kernel(const float* ws, float* out) {
    out[0] = ws[0] / (ws[1] + 1e-8f);
}

extern "C" void kernel_launch(void* const* d_in, const int* in_sizes, int n_in,
                              void* d_out, int out_size, void* d_ws, size_t ws_size,
                              hipStream_t stream) {
    const float* seq      = (const float*)d_in[0];
    const float* freq     = (const float*)d_in[1];
    const float* pres     = (const float*)d_in[2];
    const float* rmask    = (const float*)d_in[3];
    const int*   perm_idx = (const int*)d_in[4];
    const float* W1       = (const float*)d_in[5];
    const float* b1       = (const float*)d_in[6];
    const float* W2       = (const float*)d_in[7];
    const float* b2       = (const float*)d_in[8];
    const float* W3       = (const float*)d_in[9];
    const float* b3       = (const float*)d_in[10];
    float* out = (float*)d_out;
    float* ws  = (float*)d_ws;

    int Btot = in_sizes[1] / 3;      // freq is (B,3)
    int grid = Btot / 128;           // 128 rows per block (8 waves x 16-row WMMA tiles)

    size_t smem = (size_t)(64 * WP_STRIDE + 8 * 16 * ASTRIDE + 12 * 128 + 8 * 16 * 12)
                  * sizeof(float);   // ~150 KB, within 320 KB WGP LDS

    ar_init_kernel<<<1, 1, 0, stream>>>(ws);
    ar_decoder_kernel<<<grid, 256, smem, stream>>>(
        seq, freq, pres, rmask, perm_idx, W1, b1, W2, b2, W3, b3, out, ws, Btot);
    ar_fin_kernel<<<1, 1, 0, stream>>>(ws, out);
}